// IIRFilter_74904229643011
// MI455X (gfx1250) — compile-verified
//
#include <hip/hip_runtime.h>

typedef float v2f __attribute__((ext_vector_type(2)));
typedef float v8f __attribute__((ext_vector_type(8)));

#define BATCH   64
#define NSAMP   131072
#define CHUNK   128
#define NCHUNK  (NSAMP / CHUNK)   // 1024

// workspace float layout
#define WS_H    0                         // h[128]  scalar impulse response
#define WS_D0   128                       // d0[128] state-injection row (s0 comp)
#define WS_D1   256                       // d1[128] state-injection row (s1 comp)
#define WS_GV   384                       // gvec[128][2]: A^{127-i} * b2
#define WS_M2   640                       // A^128 (a00,a01,a10,a11)
#define WS_PART 1024                      // partials [BATCH][NCHUNK][2]
#define WS_SST  (1024 + BATCH*NCHUNK*2)   // chunk start states [BATCH][NCHUNK][2]

// ---------------------------------------------------------------------------
// Kernel 1: build coefficient tables (single thread; ~512 tiny 2x2 ops)
// ---------------------------------------------------------------------------
__global__ void svf_tables(const float* __restrict__ g_p,
                           const float* __restrict__ twoR_p,
                           const float* __restrict__ mix_p,
                           float* __restrict__ wsf) {
    if (threadIdx.x != 0 || blockIdx.x != 0) return;
    const float g    = g_p[0];
    const float twoR = twoR_p[0];
    const float m0 = mix_p[0], m1 = mix_p[1], m2 = mix_p[2];

    const float T   = 1.0f / (1.0f + g * (g + twoR));
    const float h00 = T,       h01 = -g * T;
    const float h10 = g * T,   h11 = T * (twoR * g + 1.0f);
    const float gb0 = g * T,   gb1 = g * g * T;          // gHB
    const float a00 = 2.0f*h00 - 1.0f, a01 = 2.0f*h01;   // A = 2H - I
    const float a10 = 2.0f*h10,        a11 = 2.0f*h11 - 1.0f;
    const float b20 = 2.0f*gb0,        b21 = 2.0f*gb1;   // b2 = 2*gHB

    // y_mixed = w_x*x + w_bp*y_bp + w_lp*y_lp
    const float w_x  = m2;
    const float w_bp = twoR * (m0 - m2);
    const float w_lp = m1 - m2;

    // h[0] = w_x + w . gHB ;  h[k] = w^T H A^{k-1} b2
    wsf[WS_H + 0] = w_x + w_bp * gb0 + w_lp * gb1;
    {
        float u0 = b20, u1 = b21;                 // A^{k-1} b2
        for (int k = 1; k < CHUNK; ++k) {
            const float Hu0 = h00*u0 + h01*u1;
            const float Hu1 = h10*u0 + h11*u1;
            wsf[WS_H + k] = w_bp * Hu0 + w_lp * Hu1;
            const float n0 = a00*u0 + a01*u1;
            const float n1 = a10*u0 + a11*u1;
            u0 = n0; u1 = n1;
        }
    }
    // d_t = w^T H A^t  (row vector), t = 0..127
    {
        float r0 = w_bp*h00 + w_lp*h10;
        float r1 = w_bp*h01 + w_lp*h11;
        for (int t = 0; t < CHUNK; ++t) {
            wsf[WS_D0 + t] = r0;
            wsf[WS_D1 + t] = r1;
            const float n0 = r0*a00 + r1*a10;     // r A
            const float n1 = r0*a01 + r1*a11;
            r0 = n0; r1 = n1;
        }
    }
    // gvec[i] = A^{127-i} b2
    {
        float u0 = b20, u1 = b21;
        wsf[WS_GV + 127*2 + 0] = u0;
        wsf[WS_GV + 127*2 + 1] = u1;
        for (int i = 126; i >= 0; --i) {
            const float n0 = a00*u0 + a01*u1;
            const float n1 = a10*u0 + a11*u1;
            u0 = n0; u1 = n1;
            wsf[WS_GV + i*2 + 0] = u0;
            wsf[WS_GV + i*2 + 1] = u1;
        }
    }
    // M2 = A^128
    {
        float p00 = 1.f, p01 = 0.f, p10 = 0.f, p11 = 1.f;
        for (int k = 0; k < CHUNK; ++k) {
            const float n00 = a00*p00 + a01*p10;
            const float n01 = a00*p01 + a01*p11;
            const float n10 = a10*p00 + a11*p10;
            const float n11 = a10*p01 + a11*p11;
            p00 = n00; p01 = n01; p10 = n10; p11 = n11;
        }
        wsf[WS_M2 + 0] = p00; wsf[WS_M2 + 1] = p01;
        wsf[WS_M2 + 2] = p10; wsf[WS_M2 + 3] = p11;
    }
}

// ---------------------------------------------------------------------------
// Kernel 2: per-chunk partial state  partial_c = sum_i gvec[i] * x[c*128+i]
// one wave per (batch, chunk); lane loads 4 contiguous f32 (b128)
// ---------------------------------------------------------------------------
__global__ __launch_bounds__(256) void svf_partials(const float* __restrict__ audio,
                                                    const float* __restrict__ wsf,
                                                    float* __restrict__ part) {
    const int wid  = blockIdx.x * 8 + (threadIdx.x >> 5);
    const int lane = threadIdx.x & 31;
    const int b = wid >> 10;              // batch
    const int c = wid & (NCHUNK - 1);     // chunk

    const float4 x4 = *reinterpret_cast<const float4*>(
        audio + (size_t)b * NSAMP + (size_t)c * CHUNK + lane * 4);
    const float* gv = wsf + WS_GV;
    const int i = lane * 4;
    float p0 = gv[(i+0)*2+0]*x4.x + gv[(i+1)*2+0]*x4.y
             + gv[(i+2)*2+0]*x4.z + gv[(i+3)*2+0]*x4.w;
    float p1 = gv[(i+0)*2+1]*x4.x + gv[(i+1)*2+1]*x4.y
             + gv[(i+2)*2+1]*x4.z + gv[(i+3)*2+1]*x4.w;
    #pragma unroll
    for (int m = 16; m >= 1; m >>= 1) {
        p0 += __shfl_xor(p0, m, 32);
        p1 += __shfl_xor(p1, m, 32);
    }
    if (lane == 0) {
        part[((size_t)b * NCHUNK + c) * 2 + 0] = p0;
        part[((size_t)b * NCHUNK + c) * 2 + 1] = p1;
    }
}

// ---------------------------------------------------------------------------
// Kernel 3: inter-chunk scan  s_{c+1} = A^128 s_c + partial_c  (64 lanes)
// ---------------------------------------------------------------------------
__global__ void svf_scan(const float* __restrict__ wsf_ro, float* __restrict__ wsf) {
    const int b = threadIdx.x;
    if (b >= BATCH) return;
    const float a00 = wsf_ro[WS_M2 + 0], a01 = wsf_ro[WS_M2 + 1];
    const float a10 = wsf_ro[WS_M2 + 2], a11 = wsf_ro[WS_M2 + 3];
    const float* part = wsf_ro + WS_PART;
    float* sst = wsf + WS_SST;
    float s0 = 0.f, s1 = 0.f;
    for (int c = 0; c < NCHUNK; ++c) {
        const size_t o = ((size_t)b * NCHUNK + c) * 2;
        sst[o + 0] = s0;
        sst[o + 1] = s1;
        const float p0 = part[o + 0], p1 = part[o + 1];
        const float n0 = a00*s0 + a01*s1 + p0;
        const float n1 = a10*s0 + a11*s1 + p1;
        s0 = n0; s1 = n1;
    }
}

// ---------------------------------------------------------------------------
// Kernel 4: per-chunk outputs via f32 WMMA.
// Y[16 batch x 128 time] = X'[16 x K] * Ct[K x 128], K = 128 (+4 state rows).
// A frag: M=batch, K=time-in. B frag: Ct[i][t] = h[t-i], read from a REVERSED
// zero-padded table hrev so the (bf.x,bf.y) pair is an ascending-address
// ds_load_2addr (no register swap). State rows 128/129 come from interleaved
// dsi[2t],dsi[2t+1] (also one ascending 2addr pair). Wave tr owns time tile
// [16tr,16tr+15]; kb=0..4tr+3 with SGPR trip count -> EXEC stays all-ones.
// ---------------------------------------------------------------------------
__global__ __launch_bounds__(256) void svf_wmma(const float* __restrict__ audio,
                                                const float* __restrict__ wsf,
                                                float* __restrict__ out) {
    __shared__ float xs[132][17];   // [K row][batch], pitch 17 (bank-conflict free)
    __shared__ float hrev[256];     // hrev[k] = h[127-k] for k<128, else 0
    __shared__ float dsi[256];      // dsi[2t]=d0[t], dsi[2t+1]=d1[t]

    const int c     = blockIdx.x;        // chunk
    const int bbase = blockIdx.y * 16;   // batch tile
    const int tid   = threadIdx.x;

    if (tid < 128) {
        hrev[tid]       = wsf[WS_H + 127 - tid];
        hrev[128 + tid] = 0.0f;
        dsi[2*tid + 0]  = wsf[WS_D0 + tid];
        dsi[2*tid + 1]  = wsf[WS_D1 + tid];
    }
    // stage x: rows 0..127, coalesced along time
    #pragma unroll
    for (int r = 0; r < 8; ++r) {
        const int idx = tid + r * 256;       // 0..2047
        const int bb  = idx >> 7;            // 0..15
        const int ii  = idx & 127;           // 0..127
        xs[ii][bb] = audio[(size_t)(bbase + bb) * NSAMP + (size_t)c * CHUNK + ii];
    }
    // rows 128/129 = chunk start state; 130/131 = zero pad
    if (tid < 16) {
        const float* sst = wsf + WS_SST;
        const size_t o = ((size_t)(bbase + tid) * NCHUNK + c) * 2;
        xs[128][tid] = sst[o + 0];
        xs[129][tid] = sst[o + 1];
        xs[130][tid] = 0.f;
        xs[131][tid] = 0.f;
    }
    __syncthreads();

    // scalar (SGPR) wave id -> scalar loop bound, EXEC untouched around WMMA
    const int tr   = __builtin_amdgcn_readfirstlane(tid >> 5);
    const int lane = tid & 31;
    const int half = lane >> 4;
    const int l15  = lane & 15;
    const int t    = tr * 16 + l15;      // output time within chunk (lane's N)

    v8f acc = {};

    const int kbmax = 4 * tr + 3;        // SGPR trip count
    for (int kb = 0; kb <= kbmax; ++kb) {
        const int i0 = kb * 4 + half * 2;
        v2f af, bf;
        af.x = xs[i0    ][l15];
        af.y = xs[i0 + 1][l15];
        // Ct[i0][t] = h[t-i0] = hrev[127 - t + i0]; pair ascending: jr, jr+1
        const int jr = 127 - t + i0;     // 0..143: always in range, no guard
        bf.x = hrev[jr];
        bf.y = hrev[jr + 1];
        acc = __builtin_amdgcn_wmma_f32_16x16x4_f32(
            false, af, false, bf, (short)0, acc, false, false);
    }
    // state-injection block: K rows 128..131 -> (d0[t], d1[t], 0, 0)
    {
        const int i0 = 128 + half * 2;
        v2f af, bf;
        af.x = xs[i0    ][l15];
        af.y = xs[i0 + 1][l15];
        const float v0 = dsi[2*t + 0];   // ascending 2addr pair
        const float v1 = dsi[2*t + 1];
        bf.x = half ? 0.0f : v0;         // value select (v_cndmask), no exec mask
        bf.y = half ? 0.0f : v1;
        acc = __builtin_amdgcn_wmma_f32_16x16x4_f32(
            false, af, false, bf, (short)0, acc, false, false);
    }
    // store: C/D layout -> VGPR v: M = v + 8*half (batch), N = l15 (time)
    #pragma unroll
    for (int v = 0; v < 8; ++v) {
        const int brow = bbase + v + half * 8;
        out[(size_t)brow * NSAMP + (size_t)c * CHUNK + t] = acc[v];
    }
}

// ---------------------------------------------------------------------------
extern "C" void kernel_launch(void* const* d_in, const int* in_sizes, int n_in,
                              void* d_out, int out_size, void* d_ws, size_t ws_size,
                              hipStream_t stream) {
    const float* audio = (const float*)d_in[0];
    const float* g     = (const float*)d_in[1];
    const float* twoR  = (const float*)d_in[2];
    const float* mix   = (const float*)d_in[3];
    float* out = (float*)d_out;
    float* wsf = (float*)d_ws;

    svf_tables<<<1, 32, 0, stream>>>(g, twoR, mix, wsf);

    svf_partials<<<(BATCH * NCHUNK) / 8, 256, 0, stream>>>(audio, wsf, wsf + WS_PART);

    svf_scan<<<1, 64, 0, stream>>>(wsf, wsf);

    dim3 grid(NCHUNK, BATCH / 16);
    svf_wmma<<<grid, 256, 0, stream>>>(audio, wsf, out);
}